// GATEncoderLayer_52106543235622
// MI455X (gfx1250) — compile-verified
//
#include <hip/hip_runtime.h>
#include <stdint.h>

// GAT encoder layer for MI455X (gfx1250, wave32).
// B=8, N=256, D=256. fp32 throughout via V_WMMA_F32_16X16X4_F32 (exact math;
// ~12MB traffic vs 0.54 GFLOP -> memory bound at 23.3 TB/s, precision is free).
// A-tiles staged to LDS via GLOBAL_LOAD_ASYNC_TO_LDS_B128 (ASYNCcnt path).

#define B_ 8
#define N_ 256
#define D_ 256
#define ALPHA_ 0.2f
#define NEG_INF_ (-1e9f)
#define LDS_PITCH 260   // 256+4 floats: row m -> banks m*4 mod 64 (conflict-free),
                        // and 260*4 = 1040 B = 65*16 B keeps 16B chunks aligned.

typedef __attribute__((ext_vector_type(2))) float v2f;
typedef __attribute__((ext_vector_type(8))) float v8f;

// Async-stage one 16x256 fp32 row-tile (16 KB) global -> LDS.
// 1024 16-byte chunks; 256 threads x 4 chunks; each wave instr moves 512 B
// with zero VGPR staging. Caller must __syncthreads() after (we wait ASYNCcnt
// here so the barrier publishes completed data).
__device__ __forceinline__ void stage_tile_async(float* tile /*LDS*/,
                                                 const float* gsrc, int tid)
{
    #pragma unroll
    for (int q = 0; q < 4; ++q) {
        int chunk = tid + 256 * q;
        int row   = chunk >> 6;                     // 64 chunks per 1024B row
        int cin   = chunk & 63;
        unsigned lds_addr =
            (unsigned)(uintptr_t)&tile[row * LDS_PITCH] + cin * 16;
        unsigned long long gaddr =
            (unsigned long long)(uintptr_t)gsrc + (unsigned long long)chunk * 16u;
        asm volatile("global_load_async_to_lds_b128 %0, %1, off"
                     :: "v"(lds_addr), "v"(gaddr) : "memory");
    }
    asm volatile("s_wait_asynccnt 0x0" ::: "memory");
}

// ---------------------------------------------------------------------------
// Kernel 1: h[b] = nodes[b] @ W + bW          (WMMA f32 16x16x4)
// Block: 256 thr = 8 waves; block tile 16 rows x 256 cols
// (wave w -> cols [16w,16w+16) and [128+16w,128+16w+16), one shared A frag).
// Grid: (16 row-tiles, 8 batches).
// ---------------------------------------------------------------------------
__global__ __launch_bounds__(256) void gat_h_gemm(
    const float* __restrict__ nodes, const float* __restrict__ W,
    const float* __restrict__ bW, float* __restrict__ h)
{
    __shared__ __align__(16) float alds[16 * LDS_PITCH];
    const int t    = threadIdx.x;
    const int wave = t >> 5;
    const int lane = t & 31;
    const int half = lane >> 4;                     // ISA A/B lane-half select
    const int mn   = lane & 15;                     // row (A) / col (B,D) index
    const int b    = blockIdx.y;
    const int i0   = blockIdx.x * 16;

    stage_tile_async(alds, nodes + (size_t)b * N_ * D_ + (size_t)i0 * D_, t);
    __syncthreads();

    const int colA = wave * 16 + mn;
    const int colB = colA + 128;
    v8f acc0 = {}, acc1 = {};
    #pragma unroll 4
    for (int k0 = 0; k0 < D_; k0 += 4) {
        const int kk = k0 + 2 * half;               // K index of frag element 0
        v2f a = *(const v2f*)&alds[mn * LDS_PITCH + kk];   // 8B aligned
        v2f b0, b1;                                 // B 4x16: lane mn = col
        b0.x = W[(size_t)kk * D_ + colA];
        b0.y = W[(size_t)(kk + 1) * D_ + colA];
        b1.x = W[(size_t)kk * D_ + colB];
        b1.y = W[(size_t)(kk + 1) * D_ + colB];
        acc0 = __builtin_amdgcn_wmma_f32_16x16x4_f32(false, a, false, b0,
                                                     (short)0, acc0, false, false);
        acc1 = __builtin_amdgcn_wmma_f32_16x16x4_f32(false, a, false, b1,
                                                     (short)0, acc1, false, false);
    }

    const float biasA = bW[colA];
    const float biasB = bW[colB];
    float* Hout = h + (size_t)b * N_ * D_;
    #pragma unroll
    for (int r = 0; r < 8; ++r) {                   // D row = r + 8*half
        int row = i0 + r + 8 * half;
        Hout[(size_t)row * D_ + colA] = acc0[r] + biasA;
        Hout[(size_t)row * D_ + colB] = acc1[r] + biasB;
    }
}

// ---------------------------------------------------------------------------
// Kernel 2: e = h @ a_src ; f = h @ a_tgt + a_b   (one wave32 per row)
// ---------------------------------------------------------------------------
__global__ __launch_bounds__(256) void gat_ef(
    const float* __restrict__ h, const float* __restrict__ a_src,
    const float* __restrict__ a_tgt, const float* __restrict__ a_b,
    float* __restrict__ e, float* __restrict__ f)
{
    const int t    = threadIdx.x;
    const int lane = t & 31;
    const int row  = blockIdx.x * 8 + (t >> 5);     // flat b*N + i
    const float* hr = h + (size_t)row * D_;

    float se = 0.f, sf = 0.f;
    #pragma unroll
    for (int q = 0; q < 8; ++q) {                   // coalesced: lane + 32*q
        int k = lane + 32 * q;
        float v = hr[k];
        se += v * a_src[k];
        sf += v * a_tgt[k];
    }
    #pragma unroll
    for (int off = 16; off > 0; off >>= 1) {        // wave32 tree reduce
        se += __shfl_xor(se, off, 32);
        sf += __shfl_xor(sf, off, 32);
    }
    if (lane == 0) { e[row] = se; f[row] = sf + a_b[0]; }
}

// ---------------------------------------------------------------------------
// Kernel 3: fused masked-leakyReLU softmax + aggregation.
// Block handles 16 attn rows: softmax -> LDS, then attn(16xN) @ h(NxD) via
// WMMA f32. attn never goes to global memory.
// Block: 256 thr = 8 waves; each wave computes 2 16x16 D-tiles.
// Grid: (16 row-tiles, 8 batches).
// ---------------------------------------------------------------------------
__global__ __launch_bounds__(256) void gat_attn_agg(
    const float* __restrict__ h, const float* __restrict__ e,
    const float* __restrict__ f, const int* __restrict__ mask,
    float* __restrict__ out)
{
    __shared__ __align__(16) float attn[16 * LDS_PITCH];
    const int t  = threadIdx.x;
    const int b  = blockIdx.y;
    const int i0 = blockIdx.x * 16;

    // ---- phase 1: row softmax of leaky_relu(e_i + f_j) with mask ----
    {
        const int r = t >> 4;                       // tile row 0..15
        const int g = t & 15;                       // 16 threads per row
        const int i = i0 + r;
        const float ei = e[b * N_ + i];
        const int*   mrow = mask + ((size_t)b * N_ + i) * N_;
        const float* fb   = f + b * N_;

        float s[16];
        float mx = NEG_INF_;
        #pragma unroll
        for (int q = 0; q < 16; ++q) {
            int j = g + 16 * q;                     // coalesced across the 16 threads
            float v = ei + fb[j];
            v = fmaxf(v, ALPHA_ * v);               // leaky relu, alpha < 1
            v = (mrow[j] > 0) ? v : NEG_INF_;
            s[q] = v;
            mx = fmaxf(mx, v);
        }
        #pragma unroll
        for (int off = 8; off > 0; off >>= 1)
            mx = fmaxf(mx, __shfl_xor(mx, off, 16));

        float sum = 0.f;
        #pragma unroll
        for (int q = 0; q < 16; ++q) {
            float p = __expf(s[q] - mx);
            s[q] = p;
            sum += p;
        }
        #pragma unroll
        for (int off = 8; off > 0; off >>= 1)
            sum += __shfl_xor(sum, off, 16);

        const float inv = 1.f / sum;
        #pragma unroll
        for (int q = 0; q < 16; ++q)
            attn[r * LDS_PITCH + g + 16 * q] = s[q] * inv;
    }
    __syncthreads();

    // ---- phase 2: out[i0:i0+16][:] = attn @ h[b] ----
    const int wave = t >> 5;
    const int lane = t & 31;
    const int half = lane >> 4;
    const int mn   = lane & 15;
    const float* Hb = h + (size_t)b * N_ * D_;
    const int colA = wave * 16 + mn;
    const int colB = colA + 128;

    v8f acc0 = {}, acc1 = {};
    #pragma unroll 4
    for (int k0 = 0; k0 < N_; k0 += 4) {
        const int kk = k0 + 2 * half;
        v2f a = *(const v2f*)&attn[mn * LDS_PITCH + kk];   // 8B aligned
        v2f b0, b1;
        b0.x = Hb[(size_t)kk * D_ + colA];
        b0.y = Hb[(size_t)(kk + 1) * D_ + colA];
        b1.x = Hb[(size_t)kk * D_ + colB];
        b1.y = Hb[(size_t)(kk + 1) * D_ + colB];
        acc0 = __builtin_amdgcn_wmma_f32_16x16x4_f32(false, a, false, b0,
                                                     (short)0, acc0, false, false);
        acc1 = __builtin_amdgcn_wmma_f32_16x16x4_f32(false, a, false, b1,
                                                     (short)0, acc1, false, false);
    }

    float* Ob = out + (size_t)b * N_ * D_;
    #pragma unroll
    for (int r = 0; r < 8; ++r) {
        int row = i0 + r + 8 * half;
        Ob[(size_t)row * D_ + colA] = acc0[r];
        Ob[(size_t)row * D_ + colB] = acc1[r];
    }
}

// ---------------------------------------------------------------------------
extern "C" void kernel_launch(void* const* d_in, const int* in_sizes, int n_in,
                              void* d_out, int out_size, void* d_ws, size_t ws_size,
                              hipStream_t stream)
{
    (void)in_sizes; (void)n_in; (void)out_size; (void)ws_size;
    const float* nodes = (const float*)d_in[0];
    const int*   mask  = (const int*)d_in[1];
    // d_in[2] = node_type (int64) unused by reference math
    const float* W     = (const float*)d_in[3];
    const float* bW    = (const float*)d_in[4];
    const float* a_src = (const float*)d_in[5];
    const float* a_tgt = (const float*)d_in[6];
    const float* a_b   = (const float*)d_in[7];
    float* out = (float*)d_out;

    float* ws = (float*)d_ws;
    float* h  = ws;                               // B*N*D floats
    float* e  = ws + (size_t)B_ * N_ * D_;        // B*N floats
    float* f  = e + B_ * N_;                      // B*N floats

    dim3 g1(N_ / 16, B_);
    gat_h_gemm<<<g1, 256, 0, stream>>>(nodes, W, bW, h);

    gat_ef<<<(B_ * N_) / 8, 256, 0, stream>>>(h, a_src, a_tgt, a_b, e, f);

    dim3 g3(N_ / 16, B_);
    gat_attn_agg<<<g3, 256, 0, stream>>>(h, e, f, mask, out);
}